// InterInstanceSeparationLoss_22385369547078
// MI455X (gfx1250) — compile-verified
//
#include <hip/hip_runtime.h>
#include <hip/hip_bf16.h>
#include <math.h>

typedef __attribute__((ext_vector_type(2))) float v2f;
typedef __attribute__((ext_vector_type(8))) float v8f;

// Problem constants (from reference setup_inputs)
constexpr int Bn = 16, Nn = 16, Hh = 512, Ww = 512, Dd = 16;
constexpr int HW = Hh * Ww;                 // 262144 pixels per batch

// Pass-2 geometry
constexpr int WAVES_PER_BLOCK = 8;          // 256 threads, wave32
constexpr int ITER = 32;                    // 32-pixel groups per wave
constexpr int PX_PER_WAVE = 32 * ITER;      // 1024
constexpr int PX_PER_BLOCK = WAVES_PER_BLOCK * PX_PER_WAVE;  // 8192
constexpr int BLOCKS_PER_BATCH = HW / PX_PER_BLOCK;          // 32
constexpr int REC = 16 * 16 + 16 + 16;      // sums[16][16] + counts[16] + tot[16] = 288

// Pass-1 geometry
constexpr int P1_PIX_PER_THREAD = 16;
constexpr int P1_BLOCK = 256;
constexpr int P1_PIX_PER_BLOCK = P1_BLOCK * P1_PIX_PER_THREAD;   // 4096
constexpr int P1_BLOCKS_PER_BATCH = HW / P1_PIX_PER_BLOCK;       // 64

// Workspace layout (bytes)
constexpr size_t OFF_BITS  = 0;                                      // u16 * B*HW = 8 MiB
constexpr size_t OFF_MSUM  = OFF_BITS + (size_t)Bn * HW * 2;         // f32[256]
constexpr size_t OFF_PART  = OFF_MSUM + 256 * 4;                     // f32[512*288]
constexpr size_t OFF_BATCH = OFF_PART + (size_t)Bn * BLOCKS_PER_BATCH * REC * 4; // f32[16*288]
constexpr size_t OFF_LOSS  = OFF_BATCH + (size_t)Bn * REC * 4;       // f32[32]

__global__ void zero_f32_kernel(float* p, int n) {
    int i = blockIdx.x * blockDim.x + threadIdx.x;
    if (i < n) p[i] = 0.0f;
}

// Pass 1: read masks once; emit per-pixel 16-bit threshold mask + per-(b,n) mask sums.
__global__ void mask_bits_kernel(const float* __restrict__ masks,
                                 unsigned short* __restrict__ bits,
                                 float* __restrict__ msum) {
    const int b   = blockIdx.x / P1_BLOCKS_PER_BATCH;
    const int blk = blockIdx.x % P1_BLOCKS_PER_BATCH;
    const int tid = threadIdx.x;

    __shared__ float s_n[Nn];
    if (tid < Nn) s_n[tid] = 0.0f;
    __syncthreads();

    float lsum[Nn];
#pragma unroll
    for (int n = 0; n < Nn; ++n) lsum[n] = 0.0f;

    const size_t mbase = (size_t)b * Nn * HW;
    const size_t bbase = (size_t)b * HW;
    for (int i = 0; i < P1_PIX_PER_THREAD; ++i) {
        const int p = blk * P1_PIX_PER_BLOCK + i * P1_BLOCK + tid;
        unsigned bv = 0;
#pragma unroll
        for (int n = 0; n < Nn; ++n) {
            const float mv = masks[mbase + (size_t)n * HW + p];
            lsum[n] += mv;
            bv |= (mv > 0.5f) ? (1u << n) : 0u;
        }
        bits[bbase + p] = (unsigned short)bv;
    }

#pragma unroll
    for (int n = 0; n < Nn; ++n) atomicAdd(&s_n[n], lsum[n]);
    __syncthreads();
    if (tid < Nn) atomicAdd(&msum[b * Nn + tid], s_n[tid]);
}

// Pass 2: segment-sum via V_WMMA_F32_16X16X4_F32.
// A = one-hot(labels 1..16) 16x4, B = emb 4x16 -> D accumulates sums[label][dim].
// Second WMMA with B = ones accumulates exact counts. Per-wave totals for background.
// Labels for the two K-slots a lane feeds are packed into one dword so a single
// ds_bpermute per WMMA step broadcasts both (halves DScnt wait pressure).
__global__ void segment_wmma_kernel(const float* __restrict__ emb,
                                    const unsigned short* __restrict__ bits,
                                    const float* __restrict__ msum,
                                    float* __restrict__ partials) {
    const int b    = blockIdx.x / BLOCKS_PER_BATCH;
    const int blk  = blockIdx.x % BLOCKS_PER_BATCH;
    const int tid  = threadIdx.x;
    const int wave = tid >> 5;
    const int lane = tid & 31;
    const int half = lane >> 4;      // 0: feeds K=0,1   1: feeds K=2,3
    const int d    = lane & 15;      // dim (B column) / A row index
    const int row1 = d + 1;          // label this lane's A-row matches

    __shared__ unsigned s_validm;
    __shared__ float smem[WAVES_PER_BLOCK][REC];
    if (tid == 0) {
        unsigned v = 0;
        for (int n = 0; n < Nn; ++n)
            if (msum[b * Nn + n] > 0.0f) v |= (1u << n);
        s_validm = v;
    }
    __syncthreads();
    const unsigned validm = s_validm;

    const float* embP = emb + (size_t)b * HW * Dd;
    const unsigned short* bitsP = bits + (size_t)b * HW;

    v8f acc  = {};
    v8f accC = {};
    v2f onesB; onesB[0] = 1.0f; onesB[1] = 1.0f;
    float tot = 0.0f;

    const int pbase0 = blk * PX_PER_BLOCK + wave * PX_PER_WAVE;
    for (int it = 0; it < ITER; ++it) {
        const int pbase = pbase0 + it * 32;
        const unsigned bv = (unsigned)bitsP[pbase + lane] & validm;
        const int label = bv ? (32 - __clz(bv)) : 0;   // highest valid n -> id n+1
        // Even lane p ends up holding {label(p) , label(p+1)<<8}.
        const int packed = (label & 0xff) | (__shfl_xor(label, 1, 32) << 8);

#pragma unroll
        for (int j = 0; j < 8; ++j) {
            const int q0 = 4 * j + half * 2;   // even pixel index of this lane's K-pair
            const float b0 = embP[(size_t)(pbase + q0) * Dd + d];
            const float b1 = embP[(size_t)(pbase + q0 + 1) * Dd + d];
            const int pl = __shfl(packed, q0, 32);      // labels of q0 (lo) and q0+1 (hi)
            const int l0 = pl & 0xff;
            const int l1 = (pl >> 8) & 0xff;
            v2f Bv; Bv[0] = b0; Bv[1] = b1;
            v2f Av; Av[0] = (l0 == row1) ? 1.0f : 0.0f;
                    Av[1] = (l1 == row1) ? 1.0f : 0.0f;
            acc  = __builtin_amdgcn_wmma_f32_16x16x4_f32(false, Av, false, Bv,
                                                         (short)0, acc, false, false);
            accC = __builtin_amdgcn_wmma_f32_16x16x4_f32(false, Av, false, onesB,
                                                         (short)0, accC, false, false);
            tot += b0 + b1;
        }
    }

    // Fold the two half-wave partial totals (same dim d in lanes L and L+16).
    tot += __shfl_xor(tot, 16, 32);

    // Stage this wave's results into LDS (C/D layout: row m = j + half*8, col = d).
    float* wrec = smem[wave];
    const int mb = half << 3;
#pragma unroll
    for (int j = 0; j < 8; ++j) wrec[(mb + j) * 16 + d] = acc[j];
    if (d == 0) {
#pragma unroll
        for (int j = 0; j < 8; ++j) wrec[256 + mb + j] = accC[j];
    }
    if (half == 0) wrec[256 + 16 + d] = tot;
    __syncthreads();

    // Deterministic fixed-order combine of the 8 wave partials -> block partial.
    float* outp = partials + (size_t)blockIdx.x * REC;
    for (int e = tid; e < REC; e += blockDim.x) {
        float s = 0.0f;
#pragma unroll
        for (int w = 0; w < WAVES_PER_BLOCK; ++w) s += smem[w][e];
        outp[e] = s;
    }
}

// Pass 2b: fixed-order reduction of per-block partials -> per-batch record.
__global__ void reduce_partials_kernel(const float* __restrict__ partials,
                                       float* __restrict__ batchrec) {
    const int b = blockIdx.x;
    const int e = threadIdx.x;          // blockDim.x == REC (288)
    float s = 0.0f;
    for (int k = 0; k < BLOCKS_PER_BATCH; ++k)
        s += partials[(size_t)(b * BLOCKS_PER_BATCH + k) * REC + e];
    batchrec[(size_t)b * REC + e] = s;
}

// Pass 3: per-batch centroids + pairwise hinge loss.
__global__ void finalize_batch_kernel(const float* __restrict__ batchrec,
                                      float* __restrict__ lossv) {
    const int b   = blockIdx.x;
    const int tid = threadIdx.x;        // blockDim.x == 64
    const float* r = batchrec + (size_t)b * REC;  // sums[16][16] | counts[16] | tot[16]

    __shared__ float cent[17][16];
    __shared__ float cnts[17];
    __shared__ int   pres[17];
    __shared__ float red[64];
    __shared__ float redc[64];

    if (tid < 16) cnts[tid + 1] = r[256 + tid];
    if (tid == 16) {
        float c0 = (float)HW;
        for (int m = 0; m < 16; ++m) c0 -= r[256 + m];
        cnts[0] = c0;
    }
    __syncthreads();

    for (int e = tid; e < 17 * 16; e += 64) {
        const int mrow = e / 16, dcol = e % 16;
        float s;
        if (mrow == 0) {
            s = r[272 + dcol];                       // total embedding sum, this dim
            for (int m = 0; m < 16; ++m) s -= r[m * 16 + dcol];
        } else {
            s = r[(mrow - 1) * 16 + dcol];
        }
        cent[mrow][dcol] = s / fmaxf(cnts[mrow], 1.0f);
    }
    if (tid < 17) pres[tid] = (cnts[tid] > 0.5f) ? 1 : 0;
    __syncthreads();

    float h = 0.0f, pc = 0.0f;
    for (int pr = tid; pr < 17 * 17; pr += 64) {
        const int i = pr / 17, j = pr % 17;
        if (i < j && pres[i] && pres[j]) {
            float sq = 0.0f;
#pragma unroll
            for (int dd = 0; dd < 16; ++dd) {
                const float df = cent[i][dd] - cent[j][dd];
                sq += df * df;
            }
            h += fmaxf(2.0f - sqrtf(sq), 0.0f);
            pc += 1.0f;
        }
    }
    red[tid] = h; redc[tid] = pc;
    __syncthreads();
    for (int s = 32; s > 0; s >>= 1) {
        if (tid < s) { red[tid] += red[tid + s]; redc[tid] += redc[tid + s]; }
        __syncthreads();
    }
    if (tid == 0) {
        int k = 0;
        for (int m = 0; m < 17; ++m) k += pres[m];
        const bool vb = (k >= 2) && (redc[0] > 0.5f);
        lossv[b]      = vb ? red[0] / fmaxf(redc[0], 1.0f) : 0.0f;
        lossv[16 + b] = vb ? 1.0f : 0.0f;
    }
}

// Pass 4: combine batches -> scalar.
__global__ void final_scalar_kernel(const float* __restrict__ lossv,
                                    float* __restrict__ out) {
    float tl = 0.0f, nv = 0.0f;
    for (int b = 0; b < Bn; ++b) { tl += lossv[b]; nv += lossv[16 + b]; }
    out[0] = (nv > 0.0f) ? tl / fmaxf(nv, 1.0f) : 0.0f;
}

extern "C" void kernel_launch(void* const* d_in, const int* in_sizes, int n_in,
                              void* d_out, int out_size, void* d_ws, size_t ws_size,
                              hipStream_t stream) {
    const float* emb   = (const float*)d_in[0];   // [16,512,512,16] f32
    const float* masks = (const float*)d_in[1];   // [16,16,512,512] f32
    float* out = (float*)d_out;

    char* ws = (char*)d_ws;
    unsigned short* bits = (unsigned short*)(ws + OFF_BITS);
    float* msum  = (float*)(ws + OFF_MSUM);
    float* part  = (float*)(ws + OFF_PART);
    float* brec  = (float*)(ws + OFF_BATCH);
    float* lossv = (float*)(ws + OFF_LOSS);

    zero_f32_kernel<<<1, 256, 0, stream>>>(msum, 256);
    mask_bits_kernel<<<Bn * P1_BLOCKS_PER_BATCH, P1_BLOCK, 0, stream>>>(masks, bits, msum);
    segment_wmma_kernel<<<Bn * BLOCKS_PER_BATCH, WAVES_PER_BLOCK * 32, 0, stream>>>(
        emb, bits, msum, part);
    reduce_partials_kernel<<<Bn, REC, 0, stream>>>(part, brec);
    finalize_batch_kernel<<<Bn, 64, 0, stream>>>(brec, lossv);
    final_scalar_kernel<<<1, 1, 0, stream>>>(lossv, out);
}